// Quantize_84284438217396
// MI455X (gfx1250) — compile-verified
//
#include <hip/hip_runtime.h>

typedef float v2f __attribute__((ext_vector_type(2)));
typedef float v8f __attribute__((ext_vector_type(8)));
typedef int   v4i_vs __attribute__((vector_size(16)));   // matches builtin param

#define GROUPS 4
#define DCODE  128   // d = C / GROUPS
#define CCH    512   // channels C

// ---------------------------------------------------------------------------
// Init: halfnorm[j] = 0.5*||e_j||^2 ; zero the commit-loss accumulator
// ---------------------------------------------------------------------------
__global__ void vq_init_kernel(const float* __restrict__ embed,
                               float* __restrict__ halfnorm,
                               float* __restrict__ accum,
                               int n_embed) {
    int c = blockIdx.x * blockDim.x + threadIdx.x;
    if (c == 0) accum[0] = 0.0f;
    if (c < n_embed) {
        const float* e = embed + (size_t)c * DCODE;
        float s = 0.0f;
        for (int t = 0; t < DCODE; t += 4) {
            float4 v = *(const float4*)(e + t);
            s += v.x * v.x + v.y * v.y + v.z * v.z + v.w * v.w;
        }
        halfnorm[c] = 0.5f * s;
    }
}

// ---------------------------------------------------------------------------
// Kernel A: distance GEMM (flat[131072,128] x embed^T[128,512]) via
// V_WMMA_F32_16X16X4_F32 + per-row argmin.  argmin(dist) == argmax(x.e - hn);
// the -hn column bias is folded into the WMMA C operand.
// Block = 256 thr = 8 waves; wave handles 16 rows; codebook staged in LDS
// via CDNA5 async global->LDS loads (ASYNCcnt) when available.
// ---------------------------------------------------------------------------
__global__ void __launch_bounds__(256)
vq_argmin_kernel(const float* __restrict__ z,
                 const float* __restrict__ embed,
                 const float* __restrict__ halfnorm,
                 int* __restrict__ ind_out,
                 int n_embed) {
    extern __shared__ float smem[];            // n_embed*DCODE + n_embed floats
    float* smem_hn = smem + n_embed * DCODE;
    const int tid   = threadIdx.x;
    const int wave  = tid >> 5;
    const int lane  = tid & 31;
    const int mrow  = lane & 15;               // row (A) / col (B,C) selector
    const int khalf = lane >> 4;               // K sub-pair 0 -> k{0,1}, 1 -> k{2,3}

    // ---- stage codebook into LDS ------------------------------------------
    const int nvec = (n_embed * DCODE) >> 2;   // float4 count
#if __has_builtin(__builtin_amdgcn_global_load_async_to_lds_b128)
    for (int i = tid; i < nvec; i += 256)
        __builtin_amdgcn_global_load_async_to_lds_b128(
            (v4i_vs*)((const char*)embed + 16 * i),
            (v4i_vs*)((char*)smem + 16 * i), 0, 0);
    asm volatile("s_wait_asynccnt 0x0" ::: "memory");
#else
    for (int i = tid; i < nvec; i += 256)
        ((float4*)smem)[i] = ((const float4*)embed)[i];
#endif
    for (int i = tid; i < n_embed; i += 256)   // stage half-norms too
        smem_hn[i] = halfnorm[i];
    __syncthreads();

    const int rowBase = blockIdx.x * 128 + wave * 16;
    const float* zrow = z + (size_t)(rowBase + mrow) * DCODE + 2 * khalf;

    // A fragments for all 32 K-chunks kept in registers (64 VGPRs)
    v2f a[32];
#pragma unroll
    for (int c = 0; c < 32; ++c)
        a[c] = *(const v2f*)(zrow + 4 * c);

    float best[8];
    int   bidx[8];
#pragma unroll
    for (int r = 0; r < 8; ++r) { best[r] = -3.402823e38f; bidx[r] = 0; }

    const int ntile = n_embed >> 4;            // 32 tiles of 16 codes
#pragma unroll 1
    for (int jt = 0; jt < ntile; ++jt) {
        const int code = (jt << 4) + mrow;     // this lane's column/code
        const float* erow = smem + code * DCODE + 2 * khalf;
        const float hn = smem_hn[code];        // column bias, from LDS
        v8f acc;
#pragma unroll
        for (int r = 0; r < 8; ++r) acc[r] = -hn;   // fold -0.5||e||^2 into C
#pragma unroll
        for (int c = 0; c < 32; ++c) {
            v2f b = *(const v2f*)(erow + 4 * c);
            acc = __builtin_amdgcn_wmma_f32_16x16x4_f32(
                false, a[c], false, b, (short)0, acc, false, false);
        }
#pragma unroll
        for (int r = 0; r < 8; ++r) {
            float s = acc[r];                  // = x.e - 0.5||e||^2
            if (s > best[r]) { best[r] = s; bidx[r] = code; }
        }
    }

    // reduce across the 16 lanes holding the same rows (columns dimension)
#pragma unroll
    for (int off = 1; off < 16; off <<= 1) {
#pragma unroll
        for (int r = 0; r < 8; ++r) {
            float ov = __shfl_xor(best[r], off, 32);
            int   oi = __shfl_xor(bidx[r], off, 32);
            if (ov > best[r] || (ov == best[r] && oi < bidx[r])) {
                best[r] = ov; bidx[r] = oi;
            }
        }
    }

    // lanes 0 / 16 own rows 0..7 / 8..15 of the wave tile
    if (mrow == 0) {
        const int rb = rowBase + 8 * khalf;
#pragma unroll
        for (int r = 0; r < 8; ++r)
            ind_out[rb + r] = bidx[r];
    }
}

// ---------------------------------------------------------------------------
// Kernel B: gather z_q into LDS (accumulating commit loss), then
// out[32768,512] = z_q @ proj_w^T + b  via V_WMMA_F32_16X16X4_F32,
// with proj_b folded into the WMMA C operand.
// Block = 256 thr = 8 waves; block tile = 32 rows x 512 cols;
// wave tile = 16 rows x 128 cols (8 x v8f accumulators).
// ---------------------------------------------------------------------------
__global__ void __launch_bounds__(256)
vq_proj_kernel(const float* __restrict__ z,
               const float* __restrict__ embed,
               const float* __restrict__ proj_w,
               const float* __restrict__ proj_b,
               const int* __restrict__ ind,
               float* __restrict__ out,
               float* __restrict__ accum) {
    __shared__ float zq[32 * CCH];             // 64 KB
    const int tid   = threadIdx.x;
    const int wave  = tid >> 5;
    const int lane  = tid & 31;
    const int mrow  = lane & 15;
    const int khalf = lane >> 4;
    const int rowBlock = blockIdx.x * 32;

    // ---- stage quantized tile + commit-loss partial -----------------------
    float dsum = 0.0f;
    for (int e = tid; e < 32 * CCH; e += 256) {
        const int ri   = e >> 9;               // row within tile
        const int c    = e & (CCH - 1);        // channel
        const int rowG = rowBlock + ri;
        const int g    = c >> 7;               // group
        const int t    = c & (DCODE - 1);
        const int code = ind[rowG * GROUPS + g];
        const float v  = embed[(size_t)code * DCODE + t];
        zq[e] = v;
        const float zv = z[(size_t)rowG * CCH + c];
        const float dd = v - zv;
        dsum += dd * dd;
    }
#pragma unroll
    for (int off = 16; off >= 1; off >>= 1)
        dsum += __shfl_xor(dsum, off, 32);
    if (lane == 0) atomicAdd(accum, dsum);
    __syncthreads();

    // ---- GEMM: wave tile 16 rows x 128 cols -------------------------------
    const int rowTile = wave >> 2;             // 0..1
    const int colTile = wave & 3;              // 0..3
    const int rowOff  = rowTile * 16;
    const int colBase = colTile * 128;

    v8f acc[8];
#pragma unroll
    for (int s = 0; s < 8; ++s) {
        const float bv = proj_b[colBase + s * 16 + mrow];  // column bias -> C
#pragma unroll
        for (int r = 0; r < 8; ++r) acc[s][r] = bv;
    }

    const float* arow = zq + (rowOff + mrow) * CCH + 2 * khalf;

#pragma unroll 1
    for (int kb = 0; kb < 4; ++kb) {           // K blocks of 128
        v2f a[32];
#pragma unroll
        for (int c = 0; c < 32; ++c)
            a[c] = *(const v2f*)(arow + kb * 128 + 4 * c);
#pragma unroll
        for (int s = 0; s < 8; ++s) {          // N sub-tiles of 16
            const float* bp = proj_w
                + (size_t)(colBase + s * 16 + mrow) * CCH
                + kb * 128 + 2 * khalf;
            if (kb < 3)                        // prefetch next K block (L2-hot)
                __builtin_prefetch(bp + 128, 0, 1);
#pragma unroll
            for (int c = 0; c < 32; ++c) {
                v2f b = *(const v2f*)(bp + 4 * c);
                acc[s] = __builtin_amdgcn_wmma_f32_16x16x4_f32(
                    false, a[c], false, b, (short)0, acc[s], false, false);
            }
        }
    }

    // ---- epilogue: store (bias already in accumulators) -------------------
#pragma unroll
    for (int s = 0; s < 8; ++s) {
        const int n  = colBase + s * 16 + mrow;
        const int rb = rowBlock + rowOff + 8 * khalf;
#pragma unroll
        for (int r = 0; r < 8; ++r)
            out[(size_t)(rb + r) * CCH + n] = acc[s][r];
    }
}

// ---------------------------------------------------------------------------
// Finalize: diff = KLD_SCALE*(COMMIT+1)*mean = 12.5 * sum / count
// ---------------------------------------------------------------------------
__global__ void vq_finalize_kernel(const float* __restrict__ accum,
                                   float* __restrict__ diff_out,
                                   float inv_count) {
    if (threadIdx.x == 0)
        diff_out[0] = 12.5f * accum[0] * inv_count;
}

// ---------------------------------------------------------------------------
extern "C" void kernel_launch(void* const* d_in, const int* in_sizes, int n_in,
                              void* d_out, int out_size, void* d_ws, size_t ws_size,
                              hipStream_t stream) {
    const float* z      = (const float*)d_in[0];
    const float* embed  = (const float*)d_in[1];
    const float* proj_w = (const float*)d_in[2];
    const float* proj_b = (const float*)d_in[3];

    const long long ztotal = (long long)in_sizes[0];   // B*H*C = 16777216
    const int  n_embed = in_sizes[1] / DCODE;          // 512
    const long long nflat = ztotal / DCODE;            // 131072 group-rows
    const long long rows  = ztotal / CCH;              // 32768 token-rows

    // d_out layout: [out (ztotal f32)] [diff (1 f32)] [ind (nflat int32)]
    float* out_f   = (float*)d_out;
    float* diff_f  = out_f + ztotal;
    int*   ind_out = (int*)(diff_f + 1);

    // d_ws layout: [0] accum ; [16..16+n_embed) halfnorm
    float* accum    = (float*)d_ws;
    float* halfnorm = accum + 16;

    vq_init_kernel<<<(n_embed + 255) / 256, 256, 0, stream>>>(
        embed, halfnorm, accum, n_embed);

    const size_t ldsA = (size_t)(n_embed * DCODE + n_embed) * sizeof(float);
    vq_argmin_kernel<<<(int)(nflat / 128), 256, ldsA, stream>>>(
        z, embed, halfnorm, ind_out, n_embed);

    vq_proj_kernel<<<(int)(rows / 32), 256, 0, stream>>>(
        z, embed, proj_w, proj_b, ind_out, out_f, accum);

    vq_finalize_kernel<<<1, 32, 0, stream>>>(
        accum, diff_f, 1.0f / (float)ztotal);
}